// LocalMean_39359080301204
// MI455X (gfx1250) — compile-verified
//
#include <hip/hip_runtime.h>
#include <hip/hip_bf16.h>

// 5x5 reflect-padded box filter over (16,3,1024,1024) fp32.
// Memory-bound: 384 MB min traffic / 23.3 TB/s ~= 16.5 us floor.
// Strategy: async global->LDS staging of reflect-padded halo tiles (CDNA5
// ASYNCcnt path), separable 5-sum compute in registers, coalesced stores.

#define PATCH   5
#define PAD     2
#define IMG_H   1024
#define IMG_W   1024
#define NPLANES (16 * 3)

#define TILE_W  128
#define TILE_H  64
#define HALO_W  (TILE_W + 2 * PAD)   // 132
#define HALO_H  (TILE_H + 2 * PAD)   // 68
#define HALO_N  (HALO_W * HALO_H)    // 8976
#define NTHREADS 256
#define ROWS_PER_THREAD (TILE_H / 2) // blockDim.y == 2

#if defined(__has_builtin)
#if __has_builtin(__builtin_amdgcn_global_load_async_to_lds_b32)
#define USE_ASYNC_LDS 1
#endif
#endif

// Builtin signature (per round-1 diagnostic) takes b32 (int) pointers in the
// global / LDS address spaces.
typedef __attribute__((address_space(1))) int g_b32_t;
typedef __attribute__((address_space(3))) int l_b32_t;

__device__ __forceinline__ int reflect_idx(int p, int n) {
    // jnp 'reflect' (edge not duplicated); PAD <= n-1 so one bounce suffices.
    p = (p < 0) ? -p : p;
    p = (p >= n) ? (2 * n - 2 - p) : p;
    return p;
}

__global__ __launch_bounds__(NTHREADS)
void LocalMean_box5_kernel(const float* __restrict__ in, float* __restrict__ out) {
    __shared__ float tile[HALO_H][HALO_W];

    const int tx0   = blockIdx.x * TILE_W;
    const int ty0   = blockIdx.y * TILE_H;
    const int plane = blockIdx.z * (IMG_H * IMG_W);   // <= 48M elems, fits int

    const float* src = in + plane;
    const int tid = threadIdx.x + threadIdx.y * TILE_W;   // 0..255

    // ---- Stage reflect-padded halo tile into LDS (async global->LDS) ----
    #pragma unroll
    for (int k = 0; k < (HALO_N + NTHREADS - 1) / NTHREADS; ++k) {
        const int i = tid + k * NTHREADS;
        if (i < HALO_N) {
            const int ly = i / HALO_W;
            const int lx = i - ly * HALO_W;
            const int gy = reflect_idx(ty0 + ly - PAD, IMG_H);
            const int gx = reflect_idx(tx0 + lx - PAD, IMG_W);
            const float* gp = src + gy * IMG_W + gx;
#ifdef USE_ASYNC_LDS
            __builtin_amdgcn_global_load_async_to_lds_b32(
                (g_b32_t*)gp, (l_b32_t*)&tile[ly][lx], 0, 0);
#else
            tile[ly][lx] = *gp;
#endif
        }
    }
#ifdef USE_ASYNC_LDS
#if __has_builtin(__builtin_amdgcn_s_wait_asynccnt)
    __builtin_amdgcn_s_wait_asynccnt(0);
#else
    asm volatile("s_wait_asynccnt 0" ::: "memory");
#endif
#endif
    __syncthreads();

    // ---- Separable 5x5 sum ----
    const int tx  = threadIdx.x;                       // 0..127 (output column)
    const int ry0 = threadIdx.y * ROWS_PER_THREAD;     // 0 or 32 (tile-local row)

    // Horizontal 5-sums for the 36 halo rows this thread's outputs touch.
    float h[ROWS_PER_THREAD + 2 * PAD];
    #pragma unroll
    for (int j = 0; j < ROWS_PER_THREAD + 2 * PAD; ++j) {
        const float* r = &tile[ry0 + j][tx];
        h[j] = ((r[0] + r[1]) + (r[2] + r[3])) + r[4];
    }

    const float inv = 1.0f / (float)(PATCH * PATCH);
    float* dst = out + plane;
    #pragma unroll
    for (int r = 0; r < ROWS_PER_THREAD; ++r) {
        const float s = ((h[r] + h[r + 1]) + (h[r + 2] + h[r + 3])) + h[r + 4];
        dst[(ty0 + ry0 + r) * IMG_W + (tx0 + tx)] = s * inv;
    }
}

extern "C" void kernel_launch(void* const* d_in, const int* in_sizes, int n_in,
                              void* d_out, int out_size, void* d_ws, size_t ws_size,
                              hipStream_t stream) {
    (void)in_sizes; (void)n_in; (void)out_size; (void)d_ws; (void)ws_size;
    const float* img = (const float*)d_in[0];
    float* out = (float*)d_out;

    dim3 block(TILE_W, 2, 1);                           // 256 threads = 8 wave32
    dim3 grid(IMG_W / TILE_W, IMG_H / TILE_H, NPLANES); // 8 x 16 x 48
    LocalMean_box5_kernel<<<grid, block, 0, stream>>>(img, out);
}